// Blur_455266533538
// MI455X (gfx1250) — compile-verified
//
#include <hip/hip_runtime.h>
#include <hip/hip_bf16.h>
#include <math.h>
#include <stdint.h>

// Blur (upfirdn2d, up=down=1, pad=(2,1), 4x4 separable FIR) on (8,128,256,256) fp32.
// Memory-bound: 512MB min traffic -> ~22us at 23.3 TB/s.
// Strategy: async global->LDS b128 tile staging (CDNA5 ASYNCcnt path),
// separable 4-tap conv from LDS with b128 LDS reads and b128 global stores.

#define IMG_H 256
#define IMG_W 256
#define TILE  64
#define LROWS  67         // TILE + 3 halo rows (y0-2 .. y0+64)
#define LSTR   72         // cols: global x0-4 .. x0+67 (18 aligned float4 chunks)
#define CHUNKS 18         // float4 chunks per row
#define NCHUNK (LROWS * CHUNKS)   // 1206

typedef int v4i __attribute__((vector_size(16)));        // int4 vector
typedef __attribute__((address_space(1))) v4i as1_v4i;   // global int4
typedef __attribute__((address_space(3))) v4i as3_v4i;   // LDS int4

__device__ __forceinline__ void async_load_f32x4(const float* gsrc, float* ldst) {
#if defined(__gfx1250__) && __has_builtin(__builtin_amdgcn_global_load_async_to_lds_b128)
    __builtin_amdgcn_global_load_async_to_lds_b128(
        (as1_v4i*)(uintptr_t)gsrc,
        (as3_v4i*)(unsigned)(uintptr_t)ldst,
        0, 0);
#else
    ldst[0] = gsrc[0]; ldst[1] = gsrc[1]; ldst[2] = gsrc[2]; ldst[3] = gsrc[3];
#endif
}

__device__ __forceinline__ void wait_async_done() {
#if defined(__gfx1250__) && __has_builtin(__builtin_amdgcn_global_load_async_to_lds_b128)
#if __has_builtin(__builtin_amdgcn_s_wait_asynccnt)
    __builtin_amdgcn_s_wait_asynccnt(0);
#else
    asm volatile("s_wait_asynccnt 0" ::: "memory");
#endif
#endif
}

__global__ __launch_bounds__(256) void blur_fir4x4_kernel(
    const float* __restrict__ x,      // (planes, 256, 256)
    const float* __restrict__ kern,   // 4x4 FIR kernel (separable, symmetric)
    float* __restrict__ out)
{
    __shared__ __align__(16) float lds[LROWS * LSTR];

    const int tid   = threadIdx.x;
    const int x0    = blockIdx.x * TILE;
    const int y0    = blockIdx.y * TILE;
    const int plane = blockIdx.z;
    const long base = (long)plane * (IMG_H * IMG_W);

    // ---- Stage input tile into LDS as aligned float4 chunks (async) ----
    // LDS col c <-> global col (x0 - 4 + c). Chunk bases are multiples of 4,
    // and IMG_W % 4 == 0, so each chunk is wholly in-bounds or wholly out.
    const float* src = x + base;
    for (int i = tid; i < NCHUNK; i += 256) {
        const int r  = i / CHUNKS;
        const int c4 = i - r * CHUNKS;
        const int gr = y0 + r - 2;
        const int gc = x0 + c4 * 4 - 4;
        float* dst = &lds[r * LSTR + c4 * 4];
        if ((unsigned)gr < IMG_H && (unsigned)gc < IMG_W) {
            async_load_f32x4(src + gr * IMG_W + gc, dst);
        } else {
            *(float4*)dst = make_float4(0.f, 0.f, 0.f, 0.f);  // ds_store_b128
        }
    }
    wait_async_done();
    __syncthreads();

    // ---- Recover exact separable 1-D taps from the 2-D kernel ----
    // kern[i][j] = v[i]*v[j]; v0 = sqrt(k00) = 0.25 exactly, v[j] = k0j/v0.
    const float v0w = sqrtf(kern[0]);
    float v[4];
    v[0] = v0w;
    v[1] = kern[1] / v0w;
    v[2] = kern[2] / v0w;
    v[3] = kern[3] / v0w;

    // ---- Each thread: 4x4 output block ----
    const int tx = tid & 15;
    const int ty = tid >> 4;
    const int X0 = tx * 4;   // local output col base
    const int Y0 = ty * 4;   // local output row base

    // Output col (x0+X0+cc) needs global input cols (x0+X0+cc-2 .. +1)
    // = LDS cols (X0+cc+2 .. X0+cc+5); taps span LDS cols X0+2 .. X0+8.
    float vs[4][7];
#pragma unroll
    for (int r = 0; r < 4; ++r)
#pragma unroll
        for (int c = 0; c < 7; ++c) vs[r][c] = 0.0f;

#pragma unroll
    for (int i = 0; i < 7; ++i) {
        const float* rp = &lds[(Y0 + i) * LSTR + X0];
        const float4 a = *(const float4*)rp;          // ds_load_b128
        const float4 b = *(const float4*)(rp + 4);    // ds_load_b128
        const float4 d = *(const float4*)(rp + 8);    // only .x used
        const float in[12] = {a.x, a.y, a.z, a.w, b.x, b.y, b.z, b.w,
                              d.x, d.y, d.z, d.w};
#pragma unroll
        for (int r = 0; r < 4; ++r) {
            if (i >= r && i <= r + 3) {
                const float wv = v[i - r];
#pragma unroll
                for (int c = 0; c < 7; ++c)
                    vs[r][c] = fmaf(wv, in[c + 2], vs[r][c]);
            }
        }
    }

    float* dstp = out + base + (long)(y0 + Y0) * IMG_W + (x0 + X0);
#pragma unroll
    for (int r = 0; r < 4; ++r) {
        float4 o;
        o.x = fmaf(v[3], vs[r][3], fmaf(v[2], vs[r][2], fmaf(v[1], vs[r][1], v[0] * vs[r][0])));
        o.y = fmaf(v[3], vs[r][4], fmaf(v[2], vs[r][3], fmaf(v[1], vs[r][2], v[0] * vs[r][1])));
        o.z = fmaf(v[3], vs[r][5], fmaf(v[2], vs[r][4], fmaf(v[1], vs[r][3], v[0] * vs[r][2])));
        o.w = fmaf(v[3], vs[r][6], fmaf(v[2], vs[r][5], fmaf(v[1], vs[r][4], v[0] * vs[r][3])));
        *(float4*)(dstp + (long)r * IMG_W) = o;       // global_store_b128
    }
}

extern "C" void kernel_launch(void* const* d_in, const int* in_sizes, int n_in,
                              void* d_out, int out_size, void* d_ws, size_t ws_size,
                              hipStream_t stream) {
    const float* x    = (const float*)d_in[0];
    const float* kern = (const float*)d_in[1];
    float* out        = (float*)d_out;

    const int planes = in_sizes[0] / (IMG_H * IMG_W);   // 8*128 = 1024
    dim3 grid(IMG_W / TILE, IMG_H / TILE, planes);      // (4, 4, 1024)
    dim3 block(256);
    blur_fir4x4_kernel<<<grid, block, 0, stream>>>(x, kern, out);
}